// LMModel_LSTM_67499706024419
// MI455X (gfx1250) — compile-verified
//
#include <hip/hip_runtime.h>
#include <math.h>

// Problem constants
#define NVOC 32000
#define DIM  256
#define HID  256
#define SEQ  512
#define BAT  8
#define DEC_SZ ((size_t)SEQ * BAT * NVOC)   // 131,072,000 floats

typedef __attribute__((ext_vector_type(2))) float v2f;
typedef __attribute__((ext_vector_type(8))) float v8f;

__device__ __forceinline__ float sigm(float x) { return 1.0f / (1.0f + __expf(-x)); }

// --------------------------------------------------------------------------
// Cross-workgroup step barrier through L2 (per-step monotonic counters,
// zeroed by kernel 1 each call). Release/acquire at device (agent) scope.
// --------------------------------------------------------------------------
__device__ __forceinline__ void group_barrier(int* cnt, int target) {
    __threadfence();            // release this thread's prior global writes
    __syncthreads();            // all threads' fences done before arrival
    if (threadIdx.x == 0) {
        __hip_atomic_fetch_add(cnt, 1, __ATOMIC_RELEASE, __HIP_MEMORY_SCOPE_AGENT);
        while (__hip_atomic_load(cnt, __ATOMIC_ACQUIRE, __HIP_MEMORY_SCOPE_AGENT) < target) {
            __builtin_amdgcn_s_sleep(1);
        }
    }
    __syncthreads();
    __threadfence();            // acquire: invalidate WGP-level cache before reads
}

// --------------------------------------------------------------------------
// Kernel 1: zero step counters + Xproj = emb[input] @ w_ih0^T + b_ih0 + b_hh0
// M=4096 (s*8+b), N=1024 (gate rows), K=256.  WMMA f32 16x16x4.
// Per-wave tile: 16Mx32N (2 accumulators, shared A fragment).
// Grid: (4 n-chunks of 256, 256 m-tiles of 16), 256 threads (8 waves).
// --------------------------------------------------------------------------
__global__ void xproj_kernel(const int* __restrict__ input,
                             const float* __restrict__ emb,
                             const float* __restrict__ w_ih0,
                             const float* __restrict__ b_ih0,
                             const float* __restrict__ b_hh0,
                             float* __restrict__ xproj,
                             int* __restrict__ cntA, int* __restrict__ cntB) {
    if (blockIdx.x == 0 && blockIdx.y == 0) {
        for (int i = threadIdx.x; i < 1024; i += 256) {
            if (i < 512) cntA[i] = 0; else cntB[i - 512] = 0;
        }
    }
    const int wave = threadIdx.x >> 5;
    const int lane = threadIdx.x & 31;
    const int lr = lane & 15;      // row (A) / col (B) within tile
    const int hi = lane >> 4;      // lane-half selects K pair / M half
    const int m0 = blockIdx.y * 16;
    const int n0 = blockIdx.x * 256 + wave * 32;

    const int tok = input[m0 + lr];                       // gather token per A-row
    const float* arow  = emb + (size_t)tok * DIM + 2 * hi; // A: X[m][k], k=k0+2*hi+v
    const float* brow0 = w_ih0 + (size_t)(n0 + lr) * DIM + 2 * hi;
    const float* brow1 = w_ih0 + (size_t)(n0 + 16 + lr) * DIM + 2 * hi;

    v8f acc0 = {}, acc1 = {};
    #pragma unroll 8
    for (int k0 = 0; k0 < DIM; k0 += 4) {
        v2f a  = *(const v2f*)(arow + k0);
        v2f b0 = *(const v2f*)(brow0 + k0);
        v2f b1 = *(const v2f*)(brow1 + k0);
        acc0 = __builtin_amdgcn_wmma_f32_16x16x4_f32(false, a, false, b0,
                                                     (short)0, acc0, false, false);
        acc1 = __builtin_amdgcn_wmma_f32_16x16x4_f32(false, a, false, b1,
                                                     (short)0, acc1, false, false);
    }
    const float bias0 = b_ih0[n0 + lr]      + b_hh0[n0 + lr];
    const float bias1 = b_ih0[n0 + 16 + lr] + b_hh0[n0 + 16 + lr];
    #pragma unroll
    for (int r = 0; r < 8; ++r) {
        const int m = m0 + r + 8 * hi;      // C layout: vgpr r -> M = r + 8*(lane>=16)
        xproj[(size_t)m * 1024 + n0 + lr]      = acc0[r] + bias0;
        xproj[(size_t)m * 1024 + n0 + 16 + lr] = acc1[r] + bias1;
    }
}

// --------------------------------------------------------------------------
// Kernel 2: persistent recurrences. 20 WGs x 1024 threads.
//   WGs 0..3  : layer-1 chain, gate-split (256 rows each, (w_ih1+w_hh1) in LDS)
//   WGs 4..19 : layer-0 chain, 64 gate-rows each of w_hh0 in LDS, Xproj reused
// Per-step cross-WG exchange via double-buffered gate vectors in L2 +
// per-step counters (group_barrier).
// --------------------------------------------------------------------------
extern __shared__ float wlds[];     // row-padded weights: stride 257 floats

__global__ void recur_kernel(const float* __restrict__ w_ih1,
                             const float* __restrict__ w_hh1,
                             const float* __restrict__ b_ih1,
                             const float* __restrict__ b_hh1,
                             const float* __restrict__ w_hh0,
                             const float* __restrict__ xproj,
                             float* __restrict__ outs,
                             float* __restrict__ gateA,   // 2 x 1024
                             float* __restrict__ gateB,   // 2 x (8*1024)
                             int* __restrict__ cntA,
                             int* __restrict__ cntB,
                             float* __restrict__ out) {
    __shared__ float s_h[2048];     // L1: h[256]   L0: hT[k*8+b]
    __shared__ float s_c[2048];     // L1: c[256]   L0: c[b*256+j]
    __shared__ float s_part[2048];
    __shared__ float s_bias[256];

    const int tid = threadIdx.x;
    const int bid = blockIdx.x;

    if (bid < 4) {
        // ---------------- Layer 1: identical-batch 256-vector chain ----------
        const int g = bid;                           // gate block (i,f,g,o)
        for (int idx = tid; idx < 256 * 256; idx += 1024) {
            int j = idx >> 8, k = idx & 255;
            wlds[j * 257 + k] = w_ih1[(size_t)(g * 256 + j) * 256 + k]
                              + w_hh1[(size_t)(g * 256 + j) * 256 + k];
        }
        if (tid < 256) {
            s_bias[tid] = b_ih1[g * 256 + tid] + b_hh1[g * 256 + tid];
            s_h[tid] = 0.0f; s_c[tid] = 0.0f;
        }
        __syncthreads();

        for (int s = 0; s < SEQ; ++s) {
            float* gbuf = gateA + (s & 1) * 1024;
            // matvec partial: thread -> (row j, K-quarter ks)
            const int j = tid & 255, ks = tid >> 8, kb = ks * 64;
            const float* wr = wlds + j * 257 + kb;
            float p = 0.0f;
            #pragma unroll 8
            for (int k = 0; k < 64; ++k) p += wr[k] * s_h[kb + k];
            s_part[j * 4 + ks] = p;
            __syncthreads();
            if (tid < 256) {
                float a = s_bias[tid] + s_part[tid * 4] + s_part[tid * 4 + 1]
                        + s_part[tid * 4 + 2] + s_part[tid * 4 + 3];
                gbuf[g * 256 + tid] = a;
            }
            group_barrier(cntA + s, 4);
            if (tid < 256) {
                float gi = gbuf[tid],       gf = gbuf[256 + tid];
                float gg = gbuf[512 + tid], go = gbuf[768 + tid];
                float cn = sigm(gf) * s_c[tid] + sigm(gi) * tanhf(gg);
                float hn = sigm(go) * tanhf(cn);
                s_c[tid] = cn; s_h[tid] = hn;
                if (g == 0) outs[(size_t)s * HID + tid] = hn;  // unique row per step
            }
            __syncthreads();
        }
        if (g == 0 && tid < 256) {          // h_t[1], c_t[1] (broadcast over batch)
            #pragma unroll
            for (int b = 0; b < BAT; ++b) {
                out[DEC_SZ + 2048 + b * 256 + tid]        = s_h[tid];
                out[DEC_SZ + 4096 + 2048 + b * 256 + tid] = s_c[tid];
            }
        }
    } else {
        // ---------------- Layer 0: batched chain, 16-way gate-row split ------
        const int wg = bid - 4;               // 0..15
        const int j0 = wg * 64;               // this WG's gate-row slice
        for (int idx = tid; idx < 64 * 256; idx += 1024) {
            int j = idx >> 8, k = idx & 255;
            wlds[j * 257 + k] = w_hh0[(size_t)(j0 + j) * 256 + k];
        }
        for (int i = tid; i < 2048; i += 1024) { s_h[i] = 0.0f; s_c[i] = 0.0f; }
        __syncthreads();

        for (int s = 0; s < SEQ; ++s) {
            float* gbuf = gateB + (s & 1) * 8192;
            // thread -> (K-half ks, row j, batch b)
            const int j = (tid >> 3) & 63, b = tid & 7, ks = tid >> 9;
            const int kb = ks * 128;
            const float* wr = wlds + j * 257 + kb;
            const float* hr = s_h + kb * 8 + b;           // hT[k*8+b]
            float p = 0.0f;
            #pragma unroll 8
            for (int k = 0; k < 128; ++k) p += wr[k] * hr[k * 8];
            s_part[(j * 8 + b) * 2 + ks] = p;
            __syncthreads();
            if (tid < 512) {
                int jj = tid >> 3, bb = tid & 7;
                float a = s_part[tid * 2] + s_part[tid * 2 + 1]
                        + xproj[(size_t)(s * 8 + bb) * 1024 + j0 + jj];
                gbuf[bb * 1024 + j0 + jj] = a;
            }
            group_barrier(cntB + s, 16);
            // redundant full-state update (cheap: 2048 elements)
            for (int i = tid; i < 2048; i += 1024) {
                int bb = i >> 8, j2 = i & 255;
                float gi = gbuf[bb * 1024 + j2];
                float gf = gbuf[bb * 1024 + 256 + j2];
                float gg = gbuf[bb * 1024 + 512 + j2];
                float go = gbuf[bb * 1024 + 768 + j2];
                float cn = sigm(gf) * s_c[i] + sigm(gi) * tanhf(gg);
                float hn = sigm(go) * tanhf(cn);
                s_c[i] = cn;
                s_h[j2 * 8 + bb] = hn;
            }
            __syncthreads();
        }
        if (bid == 4) {                       // h_t[0], c_t[0]
            for (int i = tid; i < 2048; i += 1024) {
                int bb = i >> 8, j2 = i & 255;
                out[DEC_SZ + bb * 256 + j2]        = s_h[j2 * 8 + bb];
                out[DEC_SZ + 4096 + bb * 256 + j2] = s_c[i];
            }
        }
    }
}

// --------------------------------------------------------------------------
// Kernel 3: decoded = outs_unique @ dec_w^T + dec_b, broadcast over batch.
// M=512, N=32000, K=256. WMMA f32 16x16x4, per-wave tile 16Mx32N.
// Grid (125 n-chunks of 256, 32 m-tiles of 16) x 256 thr.
// --------------------------------------------------------------------------
__global__ void decode_kernel(const float* __restrict__ outs,
                              const float* __restrict__ dec_w,
                              const float* __restrict__ dec_b,
                              float* __restrict__ out) {
    const int wave = threadIdx.x >> 5;
    const int lane = threadIdx.x & 31;
    const int lr = lane & 15, hi = lane >> 4;
    const int m0 = blockIdx.y * 16;
    const int n0 = blockIdx.x * 256 + wave * 32;

    const float* arow  = outs + (size_t)(m0 + lr) * HID + 2 * hi;
    const float* brow0 = dec_w + (size_t)(n0 + lr) * HID + 2 * hi;
    const float* brow1 = dec_w + (size_t)(n0 + 16 + lr) * HID + 2 * hi;

    v8f acc0 = {}, acc1 = {};
    #pragma unroll 8
    for (int k0 = 0; k0 < HID; k0 += 4) {
        v2f a  = *(const v2f*)(arow + k0);
        v2f b0 = *(const v2f*)(brow0 + k0);
        v2f b1 = *(const v2f*)(brow1 + k0);
        acc0 = __builtin_amdgcn_wmma_f32_16x16x4_f32(false, a, false, b0,
                                                     (short)0, acc0, false, false);
        acc1 = __builtin_amdgcn_wmma_f32_16x16x4_f32(false, a, false, b1,
                                                     (short)0, acc1, false, false);
    }
    const float bias0 = dec_b[n0 + lr];
    const float bias1 = dec_b[n0 + 16 + lr];
    #pragma unroll
    for (int r = 0; r < 8; ++r) {
        const int sidx = m0 + r + 8 * hi;     // unique sequence row
        const float v0 = acc0[r] + bias0;
        const float v1 = acc1[r] + bias1;
        float* base = out + (size_t)sidx * BAT * NVOC + n0 + lr;
        #pragma unroll
        for (int b = 0; b < BAT; ++b) {       // batch broadcast (rows identical)
            base[(size_t)b * NVOC]      = v0;
            base[(size_t)b * NVOC + 16] = v1;
        }
    }
}

// --------------------------------------------------------------------------
extern "C" void kernel_launch(void* const* d_in, const int* in_sizes, int n_in,
                              void* d_out, int out_size, void* d_ws, size_t ws_size,
                              hipStream_t stream) {
    (void)in_sizes; (void)n_in; (void)out_size; (void)ws_size;
    const int*   input = (const int*)  d_in[0];
    const float* emb   = (const float*)d_in[1];
    const float* w_ih0 = (const float*)d_in[2];
    const float* w_hh0 = (const float*)d_in[3];
    const float* b_ih0 = (const float*)d_in[4];
    const float* b_hh0 = (const float*)d_in[5];
    const float* w_ih1 = (const float*)d_in[6];
    const float* w_hh1 = (const float*)d_in[7];
    const float* b_ih1 = (const float*)d_in[8];
    const float* b_hh1 = (const float*)d_in[9];
    const float* dec_w = (const float*)d_in[10];
    const float* dec_b = (const float*)d_in[11];
    float* out = (float*)d_out;

    // workspace layout (floats)
    float* xproj = (float*)d_ws;              // 4096*1024
    float* outs  = xproj + 4096 * 1024;       // 512*256 unique h1 rows
    float* gateA = outs + 512 * 256;          // 2*1024   (layer-1 exchange)
    float* gateB = gateA + 2 * 1024;          // 2*8192   (layer-0 exchange)
    int*   cntA  = (int*)(gateB + 2 * 8192);  // 512 step counters (L1)
    int*   cntB  = cntA + 512;                // 512 step counters (L0)

    // 1) counters zeroed + x-projection GEMM (WMMA f32)
    xproj_kernel<<<dim3(4, 256), 256, 0, stream>>>(input, emb, w_ih0, b_ih0, b_hh0,
                                                   xproj, cntA, cntB);
    // 2) persistent recurrences: 4 WGs layer-1 + 16 WGs layer-0,
    //    LDS-resident weights (max 256*257*4 = 263,168 B dynamic LDS)
    recur_kernel<<<20, 1024, 256 * 257 * sizeof(float), stream>>>(
        w_ih1, w_hh1, b_ih1, b_hh1, w_hh0, xproj, outs,
        gateA, gateB, cntA, cntB, out);
    // 3) decode GEMM (WMMA f32) + batch broadcast + bias
    decode_kernel<<<dim3(125, 32), 256, 0, stream>>>(outs, dec_w, dec_b, out);
}